// DynamicAdaptiveRegionTokenizer_4896262717508
// MI455X (gfx1250) — compile-verified
//
#include <hip/hip_runtime.h>
#include <hip/hip_bf16.h>
#include <math.h>

// ---------------------------------------------------------------------------
// DynamicAdaptiveRegionTokenizer for MI455X (gfx1250, wave32)
// Pipeline: conv-score -> per-batch geometry -> bilinear patch sample ->
//           fp32 WMMA projection GEMM (V_WMMA_F32_16X16X4_F32)
// ---------------------------------------------------------------------------

#define BDIM    32
#define CCH     3
#define HDIM    224
#define WDIM    224
#define SDIM    112     // conv output rows (stride 2)
#define THH     14      // target_h
#define NPP     196     // num_patches
#define PSZ     16
#define EDIM    768
#define KDIM    768     // C*psz*psz = 3*256

typedef __attribute__((ext_vector_type(2))) float v2f;
typedef __attribute__((ext_vector_type(8))) float v8f;

__device__ __forceinline__ float softplus_f(float v) {
    // numerically stable log(1+exp(v))
    return fmaxf(v, 0.0f) + log1pf(expf(-fabsf(v)));
}

// ---------------------------------------------------------------------------
// Kernel 1: conv3x3 stride2 SAME (pad lo=0 hi=1) + bias + relu, mean over
// (oc=64, ox=112), softplus -> score[b*112 + oy].  One block per (b, oy).
// ---------------------------------------------------------------------------
__global__ __launch_bounds__(128)
void conv_score_kernel(const float* __restrict__ x,
                       const float* __restrict__ spn_w,
                       const float* __restrict__ spn_b,
                       float* __restrict__ score)
{
    __shared__ float s_in[3 * 3 * 226];   // [ic][dy][col], cols 224/225 zero pad
    __shared__ float s_w[64 * 27];
    __shared__ float s_red[128];

    const int tid = threadIdx.x;
    const int boy = blockIdx.x;
    const int b   = boy / SDIM;
    const int oy  = boy % SDIM;

    // stage input rows 2*oy .. 2*oy+2 for 3 channels (row 224 -> zero)
    for (int i = tid; i < 3 * 3 * 226; i += 128) {
        int ic  = i / (3 * 226);
        int rem = i % (3 * 226);
        int dy  = rem / 226;
        int col = rem % 226;
        int row = 2 * oy + dy;
        float v = 0.0f;
        if (row < HDIM && col < WDIM)
            v = x[((b * CCH + ic) * HDIM + row) * WDIM + col];
        s_in[i] = v;
    }
    // stage weights (64,3,3,3)
    for (int i = tid; i < 64 * 27; i += 128) s_w[i] = spn_w[i];
    __syncthreads();

    float partial = 0.0f;
    // 64 oc * 112 ox = 7168 work items
    for (int idx = tid; idx < 64 * SDIM; idx += 128) {
        int oc = idx / SDIM;
        int ox = idx % SDIM;
        float acc = 0.0f;
        const float* wv = &s_w[oc * 27];
        #pragma unroll
        for (int ic = 0; ic < 3; ++ic) {
            #pragma unroll
            for (int dy = 0; dy < 3; ++dy) {
                const float* r = &s_in[(ic * 3 + dy) * 226 + 2 * ox];
                const float* w = &wv[ic * 9 + dy * 3];
                acc += r[0] * w[0] + r[1] * w[1] + r[2] * w[2];
            }
        }
        partial += fmaxf(acc + spn_b[oc], 0.0f);   // relu(feat + bias)
    }

    s_red[tid] = partial;
    __syncthreads();
    for (int s = 64; s > 0; s >>= 1) {
        if (tid < s) s_red[tid] += s_red[tid + s];
        __syncthreads();
    }
    if (tid == 0) {
        float mean = s_red[0] / (float)(64 * SDIM);
        score[boy] = softplus_f(mean);
    }
}

// ---------------------------------------------------------------------------
// Kernel 2: per-batch geometry (serial, negligible work).
// score -> pdf -> inv_cdf(14) -> warp pdf -> inv_cdf(196) -> patch rects.
// rects[(b*196+p)*4] = {y0, y1, x0, x1}
// ---------------------------------------------------------------------------
__device__ void inv_cdf_dev(const float* cdf, int S, int n, float* out)
{
    out[0] = 0.0f;
    out[n] = (float)S;
    int j = 0;
    for (int k = 1; k < n; ++k) {
        float q = (float)k / (float)n;
        while (j < S - 1 && cdf[j + 1] < q) ++j;
        float d = cdf[j + 1] - cdf[j];
        float f = (d > 0.0f) ? (q - cdf[j]) / d : 0.0f;
        out[k] = (float)j + f;
    }
}

__global__ __launch_bounds__(32)
void geometry_kernel(const float* __restrict__ score,
                     float* __restrict__ rects)
{
    if (threadIdx.x != 0) return;
    const int b = blockIdx.x;
    const float total = (float)HDIM * 0.5f;   // 112

    float pdf[SDIM], cdf[SDIM + 1];
    const float* sc = score + b * SDIM;

    float s = 0.0f;
    for (int i = 0; i < SDIM; ++i) s += sc[i];
    for (int i = 0; i < SDIM; ++i) pdf[i] = sc[i] / s;

    cdf[0] = 0.0f;
    for (int i = 0; i < SDIM; ++i) cdf[i + 1] = cdf[i] + pdf[i];
    float cN = cdf[SDIM];
    for (int i = 0; i <= SDIM; ++i) cdf[i] /= cN;

    // row bands
    float bpos[THH + 1];
    inv_cdf_dev(cdf, SDIM, THH, bpos);
    float rh[THH];
    for (int t = 0; t < THH; ++t)
        rh[t] = (bpos[t + 1] - bpos[t]) * (total / (float)SDIM);

    // warp pdf through variable-height bands
    float cum[THH + 1];
    cum[0] = 0.0f;
    for (int t = 0; t < THH; ++t) cum[t + 1] = cum[t] + rh[t];

    const int m = SDIM / THH;   // 8
    float warped[SDIM];
    for (int t = 0; t < THH; ++t) {
        for (int j = 0; j < m; ++j) {
            float pos = cum[t] + rh[t] * (((float)j + 0.5f) / (float)m);
            float idx = pos / total * (float)SDIM;
            float u = idx - 0.5f;
            float val;
            if (u <= 0.0f)                  val = pdf[0];
            else if (u >= (float)(SDIM - 1)) val = pdf[SDIM - 1];
            else {
                int i0 = (int)floorf(u);
                float f = u - (float)i0;
                val = pdf[i0] + f * (pdf[i0 + 1] - pdf[i0]);
            }
            warped[t * m + j] = val;
        }
    }
    float ws = 0.0f;
    for (int i = 0; i < SDIM; ++i) ws += warped[i];
    for (int i = 0; i < SDIM; ++i) warped[i] /= ws;

    // second cdf + 196 patch edges
    cdf[0] = 0.0f;
    for (int i = 0; i < SDIM; ++i) cdf[i + 1] = cdf[i] + warped[i];
    cN = cdf[SDIM];
    for (int i = 0; i <= SDIM; ++i) cdf[i] /= cN;

    float edges[NPP + 1];
    inv_cdf_dev(cdf, SDIM, NPP, edges);
    // edges * (H*target_h)/edges[0,-1]; last edge == S for every batch
    const float scale = ((float)HDIM * (float)THH) / (float)SDIM;  // 28
    for (int k = 0; k <= NPP; ++k) edges[k] *= scale;

    // vertical cumsum of heights2 = rh*2
    float vcum[THH + 1];
    vcum[0] = 0.0f;
    for (int t = 0; t < THH; ++t) vcum[t + 1] = vcum[t] + rh[t] * 2.0f;

    const float rowW = (float)WDIM;
    for (int p = 0; p < NPP; ++p) {
        float e0 = edges[p], e1 = edges[p + 1];
        float c = 0.5f * (e0 + e1);
        int r = (int)floorf(c / rowW);
        r = min(max(r, 0), THH - 1);
        float x0 = fminf(fmaxf(e0 - (float)r * rowW, 0.0f), rowW);
        float x1 = fminf(fmaxf(e1 - (float)r * rowW, 0.0f), rowW);
        x1 = fmaxf(x1, x0 + 0.001f);
        float y0 = vcum[r], y1 = vcum[r + 1];
        float* o = rects + (size_t)(b * NPP + p) * 4;
        o[0] = y0; o[1] = y1; o[2] = x0; o[3] = x1;
    }
}

// ---------------------------------------------------------------------------
// Kernel 3: bilinear patch sampling -> A matrix (B*196 x 768), row-major,
// column order c*256 + py*16 + px.  One block per (b,patch), 256 threads.
// ---------------------------------------------------------------------------
__global__ __launch_bounds__(256)
void sample_kernel(const float* __restrict__ x,
                   const float* __restrict__ rects,
                   float* __restrict__ A)
{
    const int rp = blockIdx.x;          // b*196 + p
    const int b  = rp / NPP;
    const float y0r = rects[rp * 4 + 0];
    const float y1r = rects[rp * 4 + 1];
    const float x0r = rects[rp * 4 + 2];
    const float x1r = rects[rp * 4 + 3];

    const int t  = threadIdx.x;         // py*16 + px
    const int py = t >> 4;
    const int px = t & 15;

    float Y = y0r + (y1r - y0r) * (((float)py + 0.5f) / (float)PSZ);
    float X = x0r + (x1r - x0r) * (((float)px + 0.5f) / (float)PSZ);

    float yy = fminf(fmaxf(Y - 0.5f, 0.0f), (float)(HDIM - 1));
    float xx = fminf(fmaxf(X - 0.5f, 0.0f), (float)(WDIM - 1));
    int y0i = (int)floorf(yy);
    int x0i = (int)floorf(xx);
    int y1i = min(y0i + 1, HDIM - 1);
    int x1i = min(x0i + 1, WDIM - 1);
    float wy = yy - (float)y0i;
    float wx = xx - (float)x0i;

    const float* xb = x + (size_t)b * CCH * HDIM * WDIM;
    #pragma unroll
    for (int c = 0; c < CCH; ++c) {
        const float* img = xb + (size_t)c * HDIM * WDIM;
        float v00 = img[y0i * WDIM + x0i];
        float v01 = img[y0i * WDIM + x1i];
        float v10 = img[y1i * WDIM + x0i];
        float v11 = img[y1i * WDIM + x1i];
        float v = (v00 * (1.0f - wx) + v01 * wx) * (1.0f - wy) +
                  (v10 * (1.0f - wx) + v11 * wx) * wy;
        A[(size_t)rp * KDIM + c * (PSZ * PSZ) + t] = v;
    }
}

// ---------------------------------------------------------------------------
// Kernel 4: fp32 WMMA GEMM  out[M x N] = A[M x K] @ W[N x K]^T + bias
// M = B*196, N = K = 768.  One wave per 16(M) x 64(N) tile,
// V_WMMA_F32_16X16X4_F32, A fragment reused across 4 N-subtiles.
//
// fp32 16x16x4 fragment layout (ISA 7.12.2):
//   A: lane L holds row M=L&15; VGPR0/1 = K {0,1} (lanes 0-15) / {2,3} (16-31)
//   B: lane L holds col N=L&15; same K split  -> per-lane contiguous v2f load
//   C/D: VGPR v: lanes 0-15 -> M=v, N=lane; lanes 16-31 -> M=v+8, N=lane-16
// ---------------------------------------------------------------------------
__global__ __launch_bounds__(32)
void wmma_gemm_kernel(const float* __restrict__ A,
                      const float* __restrict__ Wm,     // (768, 768) row-major in K
                      const float* __restrict__ bias,
                      float* __restrict__ out)
{
    const int m0   = blockIdx.x * 16;
    const int n0   = blockIdx.y * 64;
    const int lane = threadIdx.x;
    const int half = lane >> 4;        // 0: K+0, 1: K+2
    const int lm   = lane & 15;
    const int kofs = half * 2;

    const float* aPtr = A  + (size_t)(m0 + lm) * KDIM + kofs;
    const float* bPtr = Wm + (size_t)(n0 + lm) * KDIM + kofs;

    v8f acc[4] = {v8f{}, v8f{}, v8f{}, v8f{}};

    #pragma unroll 4
    for (int k = 0; k < KDIM; k += 4) {
        v2f a = *(const v2f*)(aPtr + k);
        #pragma unroll
        for (int j = 0; j < 4; ++j) {
            v2f bb = *(const v2f*)(bPtr + (size_t)j * 16 * KDIM + k);
            acc[j] = __builtin_amdgcn_wmma_f32_16x16x4_f32(
                         false, a, false, bb, (short)0, acc[j], false, false);
        }
    }

    const int mbase = m0 + half * 8;
    #pragma unroll
    for (int j = 0; j < 4; ++j) {
        int n = n0 + j * 16 + lm;
        float bv = bias[n];
        #pragma unroll
        for (int v = 0; v < 8; ++v) {
            out[(size_t)(mbase + v) * EDIM + n] = acc[j][v] + bv;
        }
    }
}

// ---------------------------------------------------------------------------
// Launch
// ---------------------------------------------------------------------------
extern "C" void kernel_launch(void* const* d_in, const int* in_sizes, int n_in,
                              void* d_out, int out_size, void* d_ws, size_t ws_size,
                              hipStream_t stream)
{
    const float* x      = (const float*)d_in[0];
    const float* spn_w  = (const float*)d_in[1];
    const float* spn_b  = (const float*)d_in[2];
    const float* proj_w = (const float*)d_in[3];
    const float* proj_b = (const float*)d_in[4];
    // d_in[5] = target_h (14), d_in[6] = num_patches (196): fixed by shapes.

    const int B = in_sizes[0] / (CCH * HDIM * WDIM);   // 32

    // workspace layout
    float* score = (float*)d_ws;                                   // B*112
    float* rects = (float*)((char*)d_ws + 16384);                  // B*196*4
    float* A     = (float*)((char*)d_ws + 131072);                 // (B*196) x 768

    float* out = (float*)d_out;

    // 1. conv -> row scores
    conv_score_kernel<<<dim3(B * SDIM), dim3(128), 0, stream>>>(x, spn_w, spn_b, score);

    // 2. per-batch geometry -> patch rects
    geometry_kernel<<<dim3(B), dim3(32), 0, stream>>>(score, rects);

    // 3. bilinear sampling -> A matrix
    sample_kernel<<<dim3(B * NPP), dim3(256), 0, stream>>>(x, rects, A);

    // 4. projection GEMM via fp32 WMMA: M = B*196 (divisible by 16 for B=32)
    const int M = B * NPP;
    wmma_gemm_kernel<<<dim3(M / 16, EDIM / 64), dim3(32), 0, stream>>>(A, proj_w, proj_b, out);
}